// VariationalGCNEncoder_74268574482755
// MI455X (gfx1250) — compile-verified
//
#include <hip/hip_runtime.h>
#include <hip/hip_bf16.h>

// ---------------------------------------------------------------------------
// Variational GCN encoder for gfx1250 (MI455X).
// GEMMs use V_WMMA_F32_16X16X4_F32 (full fp32 precision, CDNA5 matrix pipe),
// fully unrolled with compile-time K/NC so all addressing is immediate-offset.
// Aggregation is L2-resident gather + float atomics (the true bottleneck:
// node features ~102MB fit the 192MB L2; HBM sees only compulsory traffic).
// ---------------------------------------------------------------------------

typedef __attribute__((ext_vector_type(2))) float v2f;
typedef __attribute__((ext_vector_type(8))) float v8f;

// ---------------- degree / norm kernels ----------------

__global__ void init_deg_kernel(float* __restrict__ deg, int n) {
    int i = blockIdx.x * blockDim.x + threadIdx.x;
    if (i < n) deg[i] = 1.0f;               // self-loop contributes 1
}

__global__ void count_deg_kernel(const long long* __restrict__ dst,
                                 float* __restrict__ deg, long long E) {
    long long e = (long long)blockIdx.x * blockDim.x + threadIdx.x;
    if (e < E) atomicAdd(&deg[(int)dst[e]], 1.0f);
}

__global__ void rsqrt_kernel(float* __restrict__ deg, int n) {
    int i = blockIdx.x * blockDim.x + threadIdx.x;
    if (i < n) {
        float d = deg[i];
        deg[i] = d > 0.0f ? rsqrtf(d) : 0.0f;   // in-place -> dinv
    }
}

// ---------------- fp32 WMMA GEMM: C[NR x NC] = A[NR x K] @ W[K x NC] ------
// One wave computes CT adjacent 16x16 tiles (A fragment reused CT times).
// Block = 256 threads = 8 waves covering 8*CT*16 columns; grid.x = NR/16.
//
// Fragment layouts per CDNA5 ISA 7.12.2 (32-bit, wave32):
//   A 16x4 : lanes 0-15 -> M=lane, VGPR0=K0,VGPR1=K1 ; lanes 16-31 -> K2,K3
//   B 4x16 : lanes 0-15 -> N=lane, VGPR0=K0,VGPR1=K1 ; lanes 16-31 -> K2,K3
//   C 16x16: VGPR r -> M=r (lanes 0-15), M=r+8 (lanes 16-31), N=lane&15
template<int K, int NC, int CT>
__global__ void gemm_wmma_f32(const float* __restrict__ A,
                              const float* __restrict__ W,
                              float* __restrict__ C) {
    const int rb   = blockIdx.x * 16;
    const int wave = threadIdx.x >> 5;
    const int lane = threadIdx.x & 31;
    const int cb   = wave * (16 * CT);

    const int half = lane >> 4;   // 0: K+0/K+1, 1: K+2/K+3
    const int l    = lane & 15;

    const float* __restrict__ arow = A + (size_t)(rb + l) * K;
    const float* __restrict__ wcol = W + (size_t)(2 * half) * NC + cb + l;

    // Branchless prefetch of the 1KB A row into near caches.
    __builtin_prefetch(arow, 0, 3);
    __builtin_prefetch(arow + 128, 0, 3);

    v8f acc[CT];
#pragma unroll
    for (int t = 0; t < CT; ++t) acc[t] = (v8f){};

#pragma unroll
    for (int k0 = 0; k0 < K; k0 += 4) {
        const int ka = k0 + 2 * half;
        v2f a;
        a.x = arow[ka];                       // float2 -> global_load_b64
        a.y = arow[ka + 1];
#pragma unroll
        for (int t = 0; t < CT; ++t) {
            v2f b;
            b.x = wcol[(size_t)k0 * NC + t * 16];            // K = ka
            b.y = wcol[(size_t)(k0 + 1) * NC + t * 16];      // K = ka+1
            // (neg_a, A, neg_b, B, c_mod, C, reuse_a, reuse_b)
            acc[t] = __builtin_amdgcn_wmma_f32_16x16x4_f32(
                false, a, false, b, (short)0, acc[t], false, false);
        }
    }

    const int row0 = rb + half * 8;
#pragma unroll
    for (int t = 0; t < CT; ++t)
#pragma unroll
        for (int r = 0; r < 8; ++r)
            C[(size_t)(row0 + r) * NC + cb + t * 16 + l] = acc[t][r];
}

// ---------------- self-loop + bias init: agg = b + dinv^2 * Hlin ----------

__global__ void self_plus_bias_kernel(const float* __restrict__ Hlin,
                                      const float* __restrict__ dinv,
                                      const float* __restrict__ bias,
                                      float* __restrict__ agg,
                                      int Fd4, int total) {
    int t = blockIdx.x * blockDim.x + threadIdx.x;
    if (t >= total) return;
    int i = t / Fd4;
    int c = t - i * Fd4;
    float di = dinv[i];
    float w  = di * di;
    float4 h = ((const float4*)Hlin)[(size_t)i * Fd4 + c];
    float4 b = ((const float4*)bias)[c];
    float4 o;
    o.x = fmaf(h.x, w, b.x);
    o.y = fmaf(h.y, w, b.y);
    o.z = fmaf(h.z, w, b.z);
    o.w = fmaf(h.w, w, b.w);
    ((float4*)agg)[(size_t)i * Fd4 + c] = o;
}

// ---------------- edge aggregation: agg[dst] += norm * Hlin[src] ----------
// fshift = log2(F/4). Each thread handles one float4 of one edge.

__global__ void edge_agg_kernel(const float* __restrict__ Hlin,
                                const float* __restrict__ dinv,
                                const long long* __restrict__ src,
                                const long long* __restrict__ dst,
                                float* __restrict__ agg,
                                int fshift, long long E) {
    const int Fd4 = 1 << fshift;
    const int epb = blockDim.x >> fshift;   // edges per block
    long long e = (long long)blockIdx.x * epb + (threadIdx.x >> fshift);
    if (e >= E) return;
    const int c = threadIdx.x & (Fd4 - 1);

    const int s = (int)src[e];
    const int d = (int)dst[e];
    const float nrm = dinv[s] * dinv[d];

    float4 h = ((const float4*)Hlin)[(size_t)s * Fd4 + c];
    float* out = agg + ((size_t)d * Fd4 + c) * 4;
    atomicAdd(out + 0, h.x * nrm);
    atomicAdd(out + 1, h.y * nrm);
    atomicAdd(out + 2, h.z * nrm);
    atomicAdd(out + 3, h.w * nrm);
}

// ---------------- row L2-normalize + ReLU (one wave32 per node, F=256) ----

__global__ void l2norm_relu_kernel(float* __restrict__ H, int n) {
    const int node = blockIdx.x * (blockDim.x >> 5) + (threadIdx.x >> 5);
    if (node >= n) return;
    const int lane = threadIdx.x & 31;
    float* row = H + (size_t)node * 256;

    float v[8];
    float ss = 0.0f;
#pragma unroll
    for (int j = 0; j < 8; ++j) {
        v[j] = row[lane + j * 32];
        ss = fmaf(v[j], v[j], ss);
    }
#pragma unroll
    for (int off = 16; off > 0; off >>= 1)
        ss += __shfl_xor(ss, off, 32);

    const float scale = 1.0f / fmaxf(sqrtf(ss), 1e-12f);
#pragma unroll
    for (int j = 0; j < 8; ++j) {
        float x = v[j] * scale;
        row[lane + j * 32] = x > 0.0f ? x : 0.0f;   // ReLU
    }
}

// ---------------------------------------------------------------------------

extern "C" void kernel_launch(void* const* d_in, const int* in_sizes, int n_in,
                              void* d_out, int out_size, void* d_ws, size_t ws_size,
                              hipStream_t stream) {
    // Inputs (setup_inputs order): x, edge_index, W1, b1, W_mu, b_mu, W_ls, b_ls
    const float*     x    = (const float*)d_in[0];
    const long long* ei   = (const long long*)d_in[1];   // int64 [2, E]
    const float*     W1   = (const float*)d_in[2];
    const float*     b1   = (const float*)d_in[3];
    const float*     Wmu  = (const float*)d_in[4];
    const float*     bmu  = (const float*)d_in[5];
    const float*     Wls  = (const float*)d_in[6];
    const float*     bls  = (const float*)d_in[7];

    const int       IN_C = 256, HID = 256, OUT = 128;
    const int       N    = in_sizes[0] / IN_C;           // 100000
    const long long E    = (long long)(in_sizes[1] / 2); // 3200000
    const long long* src = ei;
    const long long* dst = ei + E;

    // Workspace layout (floats): dinv[N] | Hlin[N*256] | agg/h[N*256]
    float* dinv = (float*)d_ws;
    float* Hlin = dinv + N;
    float* agg  = Hlin + (size_t)N * HID;
    // After layer 1 is aggregated, Hlin is dead -> reuse for mu/ls lin outputs.
    float* MuLin = Hlin;
    float* LsLin = Hlin + (size_t)N * OUT;

    float* mu_out = (float*)d_out;
    float* ls_out = mu_out + (size_t)N * OUT;

    // ---- degree + symmetric norm ----
    init_deg_kernel<<<(N + 255) / 256, 256, 0, stream>>>(dinv, N);
    count_deg_kernel<<<(int)((E + 255) / 256), 256, 0, stream>>>(dst, dinv, E);
    rsqrt_kernel<<<(N + 255) / 256, 256, 0, stream>>>(dinv, N);

    // ---- layer 1: Hlin = x @ W1  (K=256, NC=256, 2 col tiles/wave) ----
    gemm_wmma_f32<256, 256, 2><<<dim3(N / 16, 1), 256, 0, stream>>>(x, W1, Hlin);

    // agg = b1 + dinv^2 * Hlin, then scatter-add edges
    {
        int Fd4 = HID / 4, total = N * Fd4;
        self_plus_bias_kernel<<<(total + 255) / 256, 256, 0, stream>>>(
            Hlin, dinv, b1, agg, Fd4, total);
        int fshift = 6;                       // 64 float4 lanes per edge
        int epb = 256 >> fshift;              // 4 edges per block
        edge_agg_kernel<<<(int)((E + epb - 1) / epb), 256, 0, stream>>>(
            Hlin, dinv, src, dst, agg, fshift, E);
    }

    // ---- F.normalize(p=2) + ReLU, in place on agg -> h ----
    l2norm_relu_kernel<<<(N + 7) / 8, 256, 0, stream>>>(agg, N);

    // ---- layer 2: mu / logstd linear transforms (K=256, NC=128) ----
    gemm_wmma_f32<256, 128, 1><<<dim3(N / 16, 1), 256, 0, stream>>>(agg, Wmu, MuLin);
    gemm_wmma_f32<256, 128, 1><<<dim3(N / 16, 1), 256, 0, stream>>>(agg, Wls, LsLin);

    // ---- aggregate directly into d_out ----
    {
        int Fd4 = OUT / 4, total = N * Fd4;
        int fshift = 5;                       // 32 float4 lanes per edge
        int epb = 256 >> fshift;              // 8 edges per block

        self_plus_bias_kernel<<<(total + 255) / 256, 256, 0, stream>>>(
            MuLin, dinv, bmu, mu_out, Fd4, total);
        edge_agg_kernel<<<(int)((E + epb - 1) / epb), 256, 0, stream>>>(
            MuLin, dinv, src, dst, mu_out, fshift, E);

        self_plus_bias_kernel<<<(total + 255) / 256, 256, 0, stream>>>(
            LsLin, dinv, bls, ls_out, Fd4, total);
        edge_agg_kernel<<<(int)((E + epb - 1) / epb), 256, 0, stream>>>(
            LsLin, dinv, src, dst, ls_out, fshift, E);
    }
}